// GumbelSoftmax_7507602833465
// MI455X (gfx1250) — compile-verified
//
#include <hip/hip_runtime.h>
#include <stdint.h>

#define V_VOX 500000
#define CHUNK 4096            // floats staged per block by the TDM in pass 1
#define K1_THREADS 256
#define STREAM_BLOCKS 2048
#define STREAM_THREADS 256

typedef __attribute__((ext_vector_type(4))) unsigned int u32x4;
typedef __attribute__((ext_vector_type(4))) int i32x4;
typedef __attribute__((ext_vector_type(8))) int i32x8;
typedef __attribute__((ext_vector_type(4))) float f32x4;

// ---------------------------------------------------------------------------
// Counter-based gumbel noise: u in (0,1) from a PCG-style hash, g = -log(-log u)
// ---------------------------------------------------------------------------
__device__ __forceinline__ float gumbel_noise(unsigned int i) {
    unsigned int x = i * 747796405u + 2891336453u;
    x = ((x >> ((x >> 28u) + 4u)) ^ x) * 277803737u;
    x = (x >> 22u) ^ x;
    float u = (float)(x >> 8) * 5.9604644775390625e-08f;   // [0, 1)  (2^-24 scale)
    u = fmaxf(u, 1.1754944e-38f);                          // match jnp.finfo.tiny floor
    return -__logf(-__logf(u));
}

// ---------------------------------------------------------------------------
// Tensor Data Mover: 1-D row of f32 from global -> LDS (D# per CDNA5 ISA §8)
// ---------------------------------------------------------------------------
__device__ __forceinline__ void tdm_load_row_f32(uint32_t lds_off, const float* gptr,
                                                 uint32_t tile_elems, uint32_t valid_elems) {
    uint64_t ga = (uint64_t)(uintptr_t)gptr;
    u32x4 g0;
    g0.x = 1u;                                    // count=1, user descriptor
    g0.y = lds_off;                               // LDS byte address
    g0.z = (uint32_t)ga;                          // global_addr[31:0]
    g0.w = (uint32_t)((ga >> 32) & 0x01FFFFFFu) | (2u << 30);   // addr[56:32] | type=2
    i32x8 g1;
    g1[0] = (int)(2u << 16);                                        // data_size = 4B
    g1[1] = (int)((valid_elems & 0xFFFFu) << 16);                   // tensor_dim0 lo16
    g1[2] = (int)((valid_elems >> 16) | (1u << 16));                // dim0 hi | tensor_dim1=1
    g1[3] = (int)((tile_elems & 0xFFFFu) << 16);                    // tile_dim0
    g1[4] = (int)1u;                                                // tile_dim1=1, tile_dim2=0
    g1[5] = (int)valid_elems;                                       // tensor_dim0_stride lo
    g1[6] = 0;
    g1[7] = 0;
    i32x4 gz = {0, 0, 0, 0};
#if defined(__clang_major__) && (__clang_major__ >= 23)
    i32x8 gz8 = {0, 0, 0, 0, 0, 0, 0, 0};
    __builtin_amdgcn_tensor_load_to_lds(g0, g1, gz, gz, gz8, 0);
#else
    __builtin_amdgcn_tensor_load_to_lds(g0, g1, gz, gz, 0);
#endif
}

// ---------------------------------------------------------------------------
// K0: init per-voxel tables
// ---------------------------------------------------------------------------
__global__ void k0_init(unsigned int* __restrict__ segmax,
                        unsigned int* __restrict__ argmin, int v) {
    int i = blockIdx.x * blockDim.x + threadIdx.x;
    if (i < v) {
        segmax[i] = 0u;            // bits of +0.0f; y >= 0 so uint order == float order
        argmin[i] = 0xFFFFFFFFu;   // sentinel (> any edge id)
    }
}

// ---------------------------------------------------------------------------
// K1: per-block online (max, sum-of-exp) partials over z = e + gumbel.
//     e tile staged into LDS by the TDM (async-tensor path, TENSORcnt).
// ---------------------------------------------------------------------------
__global__ void k1_partials(const float* __restrict__ e, int E,
                            float* __restrict__ blkMax, float* __restrict__ blkSum) {
    __shared__ float tile[CHUNK];
    __shared__ float smax[K1_THREADS];
    __shared__ float ssum[K1_THREADS];

    const int blk = blockIdx.x;
    const long long base = (long long)blk * CHUNK;
    int remaining = E - (int)base;
    if (remaining > CHUNK) remaining = CHUNK;

    if (threadIdx.x < 32) {        // wave 0 issues one TDM op, waits on TENSORcnt
        uint32_t lds_off = (uint32_t)(uintptr_t)(&tile[0]);
        tdm_load_row_f32(lds_off, e + base, (uint32_t)CHUNK, (uint32_t)remaining);
        __builtin_amdgcn_s_wait_tensorcnt((short)0);
    }
    __syncthreads();

    float m = -3.402823466e+38f;
    float s = 0.0f;
    for (int t = threadIdx.x; t < remaining; t += K1_THREADS) {
        float z = tile[t] + gumbel_noise((unsigned int)(base + t));
        if (z > m) { s = s * __expf(m - z) + 1.0f; m = z; }
        else       { s += __expf(z - m); }
    }
    smax[threadIdx.x] = m;
    ssum[threadIdx.x] = s;
    __syncthreads();
    for (int off = K1_THREADS / 2; off > 0; off >>= 1) {
        if (threadIdx.x < off) {
            float m1 = smax[threadIdx.x],       s1 = ssum[threadIdx.x];
            float m2 = smax[threadIdx.x + off], s2 = ssum[threadIdx.x + off];
            float M = fmaxf(m1, m2);
            smax[threadIdx.x] = M;
            ssum[threadIdx.x] = s1 * __expf(m1 - M) + s2 * __expf(m2 - M);
        }
        __syncthreads();
    }
    if (threadIdx.x == 0) { blkMax[blk] = smax[0]; blkSum[blk] = ssum[0]; }
}

// ---------------------------------------------------------------------------
// K2: combine per-block partials -> global (M, S). Single block, fixed tree
//     order => deterministic.
// ---------------------------------------------------------------------------
__global__ void k2_combine(const float* __restrict__ blkMax,
                           const float* __restrict__ blkSum,
                           int nb, float* __restrict__ gMS) {
    __shared__ float smax[256];
    __shared__ float ssum[256];
    float m = -3.402823466e+38f;
    float s = 0.0f;
    for (int i = threadIdx.x; i < nb; i += 256) {
        float m2 = blkMax[i], s2 = blkSum[i];
        float M = fmaxf(m, m2);
        s = s * __expf(m - M) + s2 * __expf(m2 - M);
        m = M;
    }
    smax[threadIdx.x] = m;
    ssum[threadIdx.x] = s;
    __syncthreads();
    for (int off = 128; off > 0; off >>= 1) {
        if ((int)threadIdx.x < off) {
            float m1 = smax[threadIdx.x],       s1 = ssum[threadIdx.x];
            float m2 = smax[threadIdx.x + off], s2 = ssum[threadIdx.x + off];
            float M = fmaxf(m1, m2);
            smax[threadIdx.x] = M;
            ssum[threadIdx.x] = s1 * __expf(m1 - M) + s2 * __expf(m2 - M);
        }
        __syncthreads();
    }
    if (threadIdx.x == 0) { gMS[0] = smax[0]; gMS[1] = ssum[0]; }
}

// ---------------------------------------------------------------------------
// K3: y = exp(z - M) / S  -> out[0:E];  per-voxel segment max of y (bits).
//     Vectorized B128 streaming loads/stores.
// ---------------------------------------------------------------------------
__global__ void k3_y_segmax(const float* __restrict__ e, const int* __restrict__ idx,
                            int E, const float* __restrict__ gMS,
                            float* __restrict__ y_out, unsigned int* __restrict__ segmax) {
    const float M = gMS[0];
    const float S = gMS[1];
    const int stride = gridDim.x * blockDim.x;
    const int tid = blockIdx.x * blockDim.x + threadIdx.x;
    const int n4 = E >> 2;

    const f32x4* __restrict__ e4 = (const f32x4*)e;
    const i32x4* __restrict__ i4 = (const i32x4*)idx;
    f32x4*       __restrict__ y4 = (f32x4*)y_out;

    for (int i = tid; i < n4; i += stride) {
        f32x4 ev = e4[i];
        i32x4 iv = i4[i];
        unsigned int b = (unsigned int)i << 2;
        f32x4 yv;
        yv.x = __expf(ev.x + gumbel_noise(b + 0u) - M) / S;
        yv.y = __expf(ev.y + gumbel_noise(b + 1u) - M) / S;
        yv.z = __expf(ev.z + gumbel_noise(b + 2u) - M) / S;
        yv.w = __expf(ev.w + gumbel_noise(b + 3u) - M) / S;
        y4[i] = yv;
        atomicMax(&segmax[iv.x], __float_as_uint(yv.x));
        atomicMax(&segmax[iv.y], __float_as_uint(yv.y));
        atomicMax(&segmax[iv.z], __float_as_uint(yv.z));
        atomicMax(&segmax[iv.w], __float_as_uint(yv.w));
    }
    for (int i = (n4 << 2) + tid; i < E; i += stride) {   // tail
        float y = __expf(e[i] + gumbel_noise((unsigned int)i) - M) / S;
        y_out[i] = y;
        atomicMax(&segmax[idx[i]], __float_as_uint(y));
    }
}

// ---------------------------------------------------------------------------
// K4: winners (y == segment max) -> min edge id per voxel. Vectorized B128.
// ---------------------------------------------------------------------------
__global__ void k4_argmin(const float* __restrict__ y, const int* __restrict__ idx,
                          int E, const unsigned int* __restrict__ segmax,
                          unsigned int* __restrict__ argmin) {
    const int stride = gridDim.x * blockDim.x;
    const int tid = blockIdx.x * blockDim.x + threadIdx.x;
    const int n4 = E >> 2;

    const f32x4* __restrict__ y4 = (const f32x4*)y;
    const i32x4* __restrict__ i4 = (const i32x4*)idx;

    for (int i = tid; i < n4; i += stride) {
        f32x4 yv = y4[i];
        i32x4 iv = i4[i];
        unsigned int b = (unsigned int)i << 2;
        if (__float_as_uint(yv.x) == segmax[iv.x]) atomicMin(&argmin[iv.x], b + 0u);
        if (__float_as_uint(yv.y) == segmax[iv.y]) atomicMin(&argmin[iv.y], b + 1u);
        if (__float_as_uint(yv.z) == segmax[iv.z]) atomicMin(&argmin[iv.z], b + 2u);
        if (__float_as_uint(yv.w) == segmax[iv.w]) atomicMin(&argmin[iv.w], b + 3u);
    }
    for (int i = (n4 << 2) + tid; i < E; i += stride) {   // tail
        int v = idx[i];
        if (__float_as_uint(y[i]) == segmax[v]) atomicMin(&argmin[v], (unsigned int)i);
    }
}

// ---------------------------------------------------------------------------
// K5: straight-through hard output: (h - y) + y  (exactly 0 for non-winners).
//     Non-temporal store (written once, never re-read; keep L2 for y/idx).
// ---------------------------------------------------------------------------
__global__ void k5_hard(const float* __restrict__ y, const int* __restrict__ idx,
                        int E, const unsigned int* __restrict__ argmin,
                        float* __restrict__ hard) {
    const int stride = gridDim.x * blockDim.x;
    const int tid = blockIdx.x * blockDim.x + threadIdx.x;
    const int n4 = E >> 2;

    const f32x4* __restrict__ y4 = (const f32x4*)y;
    const i32x4* __restrict__ i4 = (const i32x4*)idx;
    f32x4*       __restrict__ h4 = (f32x4*)hard;

    for (int i = tid; i < n4; i += stride) {
        f32x4 yv = y4[i];
        i32x4 iv = i4[i];
        unsigned int b = (unsigned int)i << 2;
        f32x4 hv;
        hv.x = ((argmin[iv.x] == b + 0u ? 1.0f : 0.0f) - yv.x) + yv.x;
        hv.y = ((argmin[iv.y] == b + 1u ? 1.0f : 0.0f) - yv.y) + yv.y;
        hv.z = ((argmin[iv.z] == b + 2u ? 1.0f : 0.0f) - yv.z) + yv.z;
        hv.w = ((argmin[iv.w] == b + 3u ? 1.0f : 0.0f) - yv.w) + yv.w;
        __builtin_nontemporal_store(hv, &h4[i]);
    }
    for (int i = (n4 << 2) + tid; i < E; i += stride) {   // tail
        float yv = y[i];
        float h = (argmin[idx[i]] == (unsigned int)i) ? 1.0f : 0.0f;
        __builtin_nontemporal_store((h - yv) + yv, &hard[i]);
    }
}

// ---------------------------------------------------------------------------
extern "C" void kernel_launch(void* const* d_in, const int* in_sizes, int n_in,
                              void* d_out, int out_size, void* d_ws, size_t ws_size,
                              hipStream_t stream) {
    const float* e   = (const float*)d_in[0];
    const int*   cei = (const int*)d_in[1];      // shape (2, E): row0 then row1
    const int    E   = in_sizes[0];
    const int*   idx = cei + E;                  // row1 = voxel ids

    float* y    = (float*)d_out;                 // out[0:E]   = y
    float* hard = y + E;                         // out[E:2E]  = y_hard

    const int nb = (E + CHUNK - 1) / CHUNK;

    // workspace layout
    float*        blkMax = (float*)d_ws;
    float*        blkSum = blkMax + nb;
    float*        gMS    = blkSum + nb;
    unsigned int* segmax = (unsigned int*)(gMS + 2);
    unsigned int* argmin = segmax + V_VOX;

    k0_init<<<(V_VOX + 255) / 256, 256, 0, stream>>>(segmax, argmin, V_VOX);
    k1_partials<<<nb, K1_THREADS, 0, stream>>>(e, E, blkMax, blkSum);
    k2_combine<<<1, 256, 0, stream>>>(blkMax, blkSum, nb, gMS);
    k3_y_segmax<<<STREAM_BLOCKS, STREAM_THREADS, 0, stream>>>(e, idx, E, gMS, y, segmax);
    k4_argmin<<<STREAM_BLOCKS, STREAM_THREADS, 0, stream>>>(y, idx, E, segmax, argmin);
    k5_hard<<<STREAM_BLOCKS, STREAM_THREADS, 0, stream>>>(y, idx, E, argmin, hard);
}